// STGNNMixer_72696616452216
// MI455X (gfx1250) — compile-verified
//
#include <hip/hip_runtime.h>
#include <hip/hip_bf16.h>
#include <math.h>

// ---------------- problem constants ----------------
#define BB   8
#define NN   1000
#define SS   96
#define FIN  8
#define HH   64
#define PP   24
#define CC   (SS*HH)               // 6144
#define HTOT ((long)BB*NN*SS*HH)   // 49,152,000 floats per h-sized buffer

typedef float v2f __attribute__((ext_vector_type(2)));
typedef float v8f __attribute__((ext_vector_type(8)));

// Native f32 WMMA: D(16x16) = A(16x4) * B(4x16) + C  — exact f32 accumulate.
__device__ __forceinline__ v8f wmma_f32(v2f a, v2f b, v8f c) {
  return __builtin_amdgcn_wmma_f32_16x16x4_f32(false, a, false, b, (short)0, c,
                                               false, false);
}

// ---------------- async global->LDS (CDNA5, ASYNCcnt) ----------------
#if defined(__has_builtin)
#if __has_builtin(__builtin_amdgcn_global_load_async_to_lds_b128) && \
    __has_builtin(__builtin_amdgcn_s_wait_asynccnt)
#define HAVE_ASYNC_LDS 1
#endif
#endif
#ifndef HAVE_ASYNC_LDS
#define HAVE_ASYNC_LDS 0
#endif

#if HAVE_ASYNC_LDS
// clang declares the builtin as: void (*)(int __vector(4) AS1*, int __vector(4) AS3*, Imm, Imm)
typedef int v4i_gcc __attribute__((vector_size(16)));
typedef __attribute__((address_space(1))) v4i_gcc* gas_v4;
typedef __attribute__((address_space(3))) v4i_gcc* las_v4;
// generic->AS1: same 64-bit VA.  generic LDS->AS3: low 32 bits are the LDS
// offset (ISA 10.2: LDS aperture address = {shared_base[63:32], offset[31:0]}).
__device__ __forceinline__ gas_v4 as_global_v4(const void* p) {
  return (gas_v4)(unsigned long long)(uintptr_t)p;
}
__device__ __forceinline__ las_v4 as_lds_v4(void* p) {
  return (las_v4)(unsigned)(uintptr_t)p;
}
#endif

__device__ __forceinline__ void async_cp_b128(const float* g, float* l) {
#if HAVE_ASYNC_LDS
  __builtin_amdgcn_global_load_async_to_lds_b128(as_global_v4(g), as_lds_v4(l),
                                                 0, 0);
#else
  l[0] = g[0]; l[1] = g[1]; l[2] = g[2]; l[3] = g[3];
#endif
}
// wait until ASYNCcnt <= n (n must be a compile-time small constant here)
__device__ __forceinline__ void wait_async(int n) {
#if HAVE_ASYNC_LDS
  if (n == 0)      __builtin_amdgcn_s_wait_asynccnt(0);
  else             __builtin_amdgcn_s_wait_asynccnt(2);
#else
  (void)n;
#endif
}

// ---------------------------------------------------------------
// K1: combined = alpha*adj + (1-alpha)*softmax(relu(E E^T)), diag=1,
//     dis[m] = rsqrt(max(rowsum,1)).  One block per row m.
// ---------------------------------------------------------------
__global__ void k_combined(const float* __restrict__ adj,
                           const float* __restrict__ emb,
                           const float* __restrict__ alpha_p,
                           float* __restrict__ comb,
                           float* __restrict__ dis) {
  __shared__ float row[NN];
  __shared__ float red[256];
  __shared__ float em[HH];
  const int m = blockIdx.x;
  const int t = threadIdx.x;
  if (t < HH) em[t] = emb[m*HH + t];
  __syncthreads();

  float lmax = -1e30f;
  for (int n = t; n < NN; n += 256) {
    const float* en = emb + n*HH;
    float acc = 0.f;
#pragma unroll
    for (int k = 0; k < HH; ++k) acc = fmaf(em[k], en[k], acc);
    acc = fmaxf(acc, 0.f);            // relu
    row[n] = acc;
    lmax = fmaxf(lmax, acc);
  }
  red[t] = lmax; __syncthreads();
  for (int s = 128; s > 0; s >>= 1) { if (t < s) red[t] = fmaxf(red[t], red[t+s]); __syncthreads(); }
  const float mx = red[0]; __syncthreads();

  float lsum = 0.f;
  for (int n = t; n < NN; n += 256) { float e = expf(row[n] - mx); row[n] = e; lsum += e; }
  red[t] = lsum; __syncthreads();
  for (int s = 128; s > 0; s >>= 1) { if (t < s) red[t] += red[t+s]; __syncthreads(); }
  const float inv = 1.0f / red[0]; __syncthreads();

  const float alpha = *alpha_p;
  float rowsum = 0.f;
  for (int n = t; n < NN; n += 256) {
    float c = alpha*adj[m*NN + n] + (1.0f - alpha)*(row[n]*inv);
    if (n == m) c = 1.0f;
    comb[m*NN + n] = c;
    rowsum += c;
  }
  red[t] = rowsum; __syncthreads();
  for (int s = 128; s > 0; s >>= 1) { if (t < s) red[t] += red[t+s]; __syncthreads(); }
  if (t == 0) dis[m] = rsqrtf(fmaxf(red[0], 1.0f));
}

// K2: A = dis[m] * comb * dis[n]
__global__ void k_gcnnorm(const float* __restrict__ comb,
                          const float* __restrict__ dis,
                          float* __restrict__ A) {
  int i = blockIdx.x*256 + threadIdx.x;
  if (i < NN*NN) {
    int m = i / NN, n = i - m*NN;
    A[i] = dis[m]*comb[i]*dis[n];
  }
}

// K3: h = x @ W_proj + b_proj + node_emb   (K=8, VALU)
__global__ void k_proj(const float* __restrict__ x,
                       const float* __restrict__ Wp,
                       const float* __restrict__ bp,
                       const float* __restrict__ emb,
                       float* __restrict__ h) {
  long r = blockIdx.x;                 // (b*N + n)*S + s
  int hc = threadIdx.x;                // 0..63
  long n = (r / SS) % NN;
  const float* xr = x + r*FIN;
  float acc = bp[hc] + emb[n*HH + hc];
#pragma unroll
  for (int k = 0; k < FIN; ++k) acc = fmaf(xr[k], Wp[k*HH + hc], acc);
  h[r*HH + hc] = acc;
}

// ---------------------------------------------------------------
// K4: temporal mixer, WMMA.  out[s',hc] = sum_s Wt[s,s'] * h[s,hc]
//     h[s',hc] += gelu(out + bt[s']).  One block per (b,n); 8 waves,
//     24 tiles.  h tile staged via async B128 copies (ASYNCcnt).
// ---------------------------------------------------------------
__global__ void k_temporal(float* __restrict__ h,
                           const float* __restrict__ Wt,
                           const float* __restrict__ bt) {
  __shared__ float hT[SS*HH];          // [s][hc], 24 KB
  float* hp = h + (long)blockIdx.x * (SS*HH);
  const int t = threadIdx.x;
#pragma unroll
  for (int r = 0; r < 6; ++r) {        // 6144 floats = 1536 x B128
    int idx = (r*256 + t) * 4;
    async_cp_b128(hp + idx, hT + idx);
  }
  wait_async(0);
  __syncthreads();

  const int wave = t >> 5, lane = t & 31;
  const int lg = lane >> 4, ll = lane & 15;
  for (int j = 0; j < 3; ++j) {
    const int tile = wave*3 + j;                // 0..23
    const int m0 = (tile >> 2) * 16;            // s' tile
    const int n0 = (tile & 3) * 16;             // hc tile
    v8f acc = {};
    for (int k0 = 0; k0 < SS; k0 += 4) {
      const int ka = k0 + lg*2;
      v2f a, b;
      a.x = Wt[(ka+0)*SS + m0 + ll];            // A[m,k] = Wt[k, m]
      a.y = Wt[(ka+1)*SS + m0 + ll];
      b.x = hT[(ka+0)*HH + n0 + ll];            // B[k,n] = h[s=k, hc=n]
      b.y = hT[(ka+1)*HH + n0 + ll];
      acc = wmma_f32(a, b, acc);
    }
#pragma unroll
    for (int r = 0; r < 8; ++r) {
      const int srow = m0 + lg*8 + r;
      float v = acc[r] + bt[srow];
      float ge = 0.5f * v * (1.0f + erff(v * 0.70710678118654752f)); // exact gelu
      hp[srow*HH + n0 + ll] += ge;
    }
  }
}

// ---------------------------------------------------------------
// K5: hw = h @ Wg  (per (b,n): 96x64x64, WMMA).  h staged transposed
//     in LDS so A-frag reads are conflict-free.
// ---------------------------------------------------------------
__global__ void k_hw(const float* __restrict__ h,
                     const float* __restrict__ Wg,
                     float* __restrict__ hw) {
  __shared__ float hTt[HH*SS];         // [hc][s]
  const long bn = blockIdx.x;
  const float* hp = h + bn*(SS*HH);
  float* hwp = hw + bn*(SS*HH);
  const int t = threadIdx.x;
  for (int idx = t; idx < SS*HH; idx += 256) {
    int srow = idx >> 6, hc = idx & 63;        // idx = srow*64 + hc
    hTt[hc*SS + srow] = hp[idx];
  }
  __syncthreads();

  const int wave = t >> 5, lane = t & 31;
  const int lg = lane >> 4, ll = lane & 15;
  for (int j = 0; j < 3; ++j) {
    const int tile = wave*3 + j;
    const int m0 = (tile >> 2) * 16;            // s tile
    const int n0 = (tile & 3) * 16;             // h' tile
    v8f acc = {};
    for (int k0 = 0; k0 < HH; k0 += 4) {
      const int ka = k0 + lg*2;
      v2f a, b;
      a.x = hTt[(ka+0)*SS + m0 + ll];           // A[m=s, k=hc]
      a.y = hTt[(ka+1)*SS + m0 + ll];
      b.x = Wg[(ka+0)*HH + n0 + ll];            // B[k=hc, n=h']
      b.y = Wg[(ka+1)*HH + n0 + ll];
      acc = wmma_f32(a, b, acc);
    }
#pragma unroll
    for (int r = 0; r < 8; ++r)
      hwp[(m0 + lg*8 + r)*HH + n0 + ll] = acc[r];
  }
}

// ---------------------------------------------------------------
// K6: g[b,m,c] = sum_n A[m,n] * hw[b,n,c]   (c = s*64+h, 6144 cols)
//     Dominant GEMM: per b, [1000x1000] x [1000x6144].
//     Block = 32 rows x 256 cols; each wave owns a 2x2 grid of
//     16x16 tiles (4 WMMAs per K-step, A/B frags each reused 2x).
//     A panel (32 x 64k) double-buffered in LDS via async B128
//     copies; prefetch of chunk k+64 overlaps WMMA on chunk k.
//     B streams from L2 (24.6 MB panel/batch << 192 MB L2).
// ---------------------------------------------------------------
#define KCH 64
#define LDA 68   // row stride: 272 B (16B-aligned rows, conflict-free frags)

__global__ void k_graph(const float* __restrict__ A,
                        const float* __restrict__ hw,
                        float* __restrict__ g) {
  __shared__ float As[2*32*LDA];       // 2 x (32 rows x 68) = 17.4 KB
  const int b  = blockIdx.z;
  const int m0 = blockIdx.y * 32;      // rows m0..m0+31 (>=NN rows discarded)
  const int t  = threadIdx.x;
  const int wave = t >> 5, lane = t & 31;
  const int lg = lane >> 4, ll = lane & 15;
  const int c0 = blockIdx.x * 256 + wave * 16;     // second N tile at c0+128
  const float* hwb = hw + (long)b * NN * CC;
  float* gb = g + (long)b * NN * CC;

  // stage one K-chunk of the A panel: 32 rows x 64 k = 512 B128 ops
  auto issue_chunk = [&](float* buf, int kk) {
#pragma unroll
    for (int r = 0; r < 2; ++r) {
      int idx = r*256 + t;             // 0..511
      int ml  = idx >> 4;              // 0..31
      int k4  = (idx & 15) << 2;       // 0,4,...,60
      int m   = m0 + ml; if (m > NN-1) m = NN-1;   // clamp: rows>=NN discarded
      async_cp_b128(A + (long)m*NN + kk + k4, buf + ml*LDA + k4);
    }
  };

  const int NCHUNK = (NN + KCH - 1) / KCH;         // 16 (last chunk: 40 valid k)
  v8f acc00 = {}, acc01 = {}, acc10 = {}, acc11 = {};

  issue_chunk(&As[0], 0);
  for (int ic = 0; ic < NCHUNK; ++ic) {
    float* cur = &As[(ic & 1) * 32 * LDA];
    const int kk = ic * KCH;
    if (ic + 1 < NCHUNK) {
      issue_chunk(&As[((ic + 1) & 1) * 32 * LDA], kk + KCH);
      wait_async(2);                   // current chunk's 2 loads complete
    } else {
      wait_async(0);
    }
    __syncthreads();

    const int kmax = (NN - kk) < KCH ? (NN - kk) : KCH;  // 64 or 40
    for (int kl = 0; kl < kmax; kl += 4) {
      const int ka = kl + lg*2;
      const long kg = kk + ka;
      v2f a0, a1, b0, b1;
      a0.x = cur[(ll)     *LDA + ka];  a0.y = cur[(ll)     *LDA + ka + 1];
      a1.x = cur[(16 + ll)*LDA + ka];  a1.y = cur[(16 + ll)*LDA + ka + 1];
      b0.x = hwb[(kg+0)*CC + c0 + ll];       b0.y = hwb[(kg+1)*CC + c0 + ll];
      b1.x = hwb[(kg+0)*CC + c0 + 128 + ll]; b1.y = hwb[(kg+1)*CC + c0 + 128 + ll];
      acc00 = wmma_f32(a0, b0, acc00);
      acc01 = wmma_f32(a0, b1, acc01);
      acc10 = wmma_f32(a1, b0, acc10);
      acc11 = wmma_f32(a1, b1, acc11);
    }
    __syncthreads();                   // all waves done before buffer reuse
  }

#pragma unroll
  for (int r = 0; r < 8; ++r) {
    const int mA = m0 + lg*8 + r;      // M tile 0
    const int mB = mA + 16;            // M tile 1
    if (mA < NN) {
      gb[(long)mA*CC + c0 + ll]       = acc00[r];
      gb[(long)mA*CC + c0 + 128 + ll] = acc01[r];
    }
    if (mB < NN) {
      gb[(long)mB*CC + c0 + ll]       = acc10[r];
      gb[(long)mB*CC + c0 + 128 + ll] = acc11[r];
    }
  }
}

// K7: h = layer_norm(h + g + bg) * ln_g + ln_b   (over H=64; 1 wave/row)
__global__ void k_ln(float* __restrict__ h, const float* __restrict__ g,
                     const float* __restrict__ bg,
                     const float* __restrict__ lng,
                     const float* __restrict__ lnb) {
  const long row = (long)blockIdx.x * 8 + (threadIdx.x >> 5);
  const int lane = threadIdx.x & 31;
  float* hp = h + row*HH;
  const float* gp = g + row*HH;
  float v0 = hp[lane]      + gp[lane]      + bg[lane];
  float v1 = hp[lane + 32] + gp[lane + 32] + bg[lane + 32];
  float s = v0 + v1;
#pragma unroll
  for (int off = 16; off > 0; off >>= 1) s += __shfl_xor(s, off, 32);
  const float mu = s * (1.0f/64.0f);
  const float d0 = v0 - mu, d1 = v1 - mu;
  float q = d0*d0 + d1*d1;
#pragma unroll
  for (int off = 16; off > 0; off >>= 1) q += __shfl_xor(q, off, 32);
  const float inv = rsqrtf(q * (1.0f/64.0f) + 1e-5f);
  hp[lane]      = d0*inv*lng[lane]      + lnb[lane];
  hp[lane + 32] = d1*inv*lng[lane + 32] + lnb[lane + 32];
}

// K8: head GEMM [8000 x 6144] x [6144 x 24], WMMA (N padded to 32).
__global__ void k_head(const float* __restrict__ h,
                       const float* __restrict__ Wh,
                       const float* __restrict__ bh,
                       float* __restrict__ out) {
  const long bn0 = (long)blockIdx.x * 16;
  const int t = threadIdx.x;
  const int wave = t >> 5, lane = t & 31;
  const int lg = lane >> 4, ll = lane & 15;
  const int p0 = wave * 16;
  const int p  = p0 + ll;
  const bool pv = (p < PP);
  v8f acc = {};
  for (int k0 = 0; k0 < CC; k0 += 4) {
    const int ka = k0 + lg*2;
    v2f a, b;
    a.x = h[(bn0 + ll)*CC + ka + 0];
    a.y = h[(bn0 + ll)*CC + ka + 1];
    b.x = pv ? Wh[(ka+0)*PP + p] : 0.0f;
    b.y = pv ? Wh[(ka+1)*PP + p] : 0.0f;
    acc = wmma_f32(a, b, acc);
  }
#pragma unroll
  for (int r = 0; r < 8; ++r) {
    const long m = bn0 + lg*8 + r;
    if (pv) out[m*PP + p] = acc[r] + bh[p];
  }
}

// ---------------------------------------------------------------
extern "C" void kernel_launch(void* const* d_in, const int* in_sizes, int n_in,
                              void* d_out, int out_size, void* d_ws, size_t ws_size,
                              hipStream_t stream) {
  const float* x     = (const float*)d_in[0];
  const float* adj   = (const float*)d_in[1];
  const float* emb   = (const float*)d_in[2];
  const float* alpha = (const float*)d_in[3];
  const float* Wp    = (const float*)d_in[4];
  const float* bp    = (const float*)d_in[5];
  const float* Wt    = (const float*)d_in[6];
  const float* bt    = (const float*)d_in[7];
  const float* Wg    = (const float*)d_in[8];
  const float* bg    = (const float*)d_in[9];
  const float* lng   = (const float*)d_in[10];
  const float* lnb   = (const float*)d_in[11];
  const float* Wh    = (const float*)d_in[12];
  const float* bh    = (const float*)d_in[13];
  float* out = (float*)d_out;

  float* ws   = (float*)d_ws;
  float* h    = ws;                    // 49.15M floats
  float* hw   = h    + HTOT;           // 49.15M
  float* g    = hw   + HTOT;           // 49.15M
  float* comb = g    + HTOT;           // 1M
  float* Amat = comb + (long)NN*NN;    // 1M  (dis must follow: K6 tail reads
  float* dis  = Amat + (long)NN*NN;    //      over-fetch <1000 floats past Amat)

  // adjacency pipeline
  k_combined<<<NN, 256, 0, stream>>>(adj, emb, alpha, comb, dis);
  k_gcnnorm<<<(NN*NN + 255)/256, 256, 0, stream>>>(comb, dis, Amat);

  // projection
  k_proj<<<(long)BB*NN*SS, HH, 0, stream>>>(x, Wp, bp, emb, h);

  for (int i = 0; i < 2; ++i) {
    k_temporal<<<BB*NN, 256, 0, stream>>>(h, Wt + (long)i*SS*SS, bt + (long)i*SS);
    k_hw<<<BB*NN, 256, 0, stream>>>(h, Wg + (long)i*HH*HH, hw);
    dim3 gg(CC/256, (NN + 31)/32, BB);            // 24 x 32 x 8
    k_graph<<<gg, 256, 0, stream>>>(Amat, hw, g);
    k_ln<<<(BB*NN*SS)/8, 256, 0, stream>>>(h, g, bg + (long)i*HH,
                                           lng + (long)i*HH, lnb + (long)i*HH);
  }
  k_head<<<(BB*NN)/16, 64, 0, stream>>>(h, Wh, bh, out);
}